// Poincare_loss_5016521802072
// MI455X (gfx1250) — compile-verified
//
#include <hip/hip_runtime.h>
#include <hip/hip_bf16.h>

// Poincare NT-Xent loss, MI455X (gfx1250, wave32).
// Gram matrix via V_WMMA_F32_16X16X4_F32 (two K=4 steps for K=8),
// elementwise arcosh/exp using bare hardware trans ops (v_sqrt/v_log/v_exp/v_rcp),
// deterministic ordered reductions.

typedef __attribute__((ext_vector_type(2))) float v2f;
typedef __attribute__((ext_vector_type(8))) float v8f;

#define NROWS    8192        // 2N
#define NTILES   512         // NROWS / 16
#define BOUNDARY (1.0f - 1e-5f)
#define LN2      0.6931471805599453f
#define TWO_OVER_LN2 2.8853900817779268f   // (1/T)/ln2, T = 0.5

// ---------------------------------------------------------------- prep ------
// Pack z = [z_i ; z_j] contiguously, precompute raw squared norms and
// 1/(1 - clamp(raw, 0, BOUNDARY)) per row.
__global__ void poincare_prep_kernel(const float* __restrict__ zi,
                                     const float* __restrict__ zj,
                                     float* __restrict__ zbuf,
                                     float* __restrict__ raw,
                                     float* __restrict__ inv1m) {
  int i = blockIdx.x * blockDim.x + threadIdx.x;
  if (i >= NROWS) return;
  const float* src = (i < NROWS / 2) ? (zi + i * 8) : (zj + (i - NROWS / 2) * 8);
  float s = 0.0f;
#pragma unroll
  for (int k = 0; k < 8; ++k) {
    float v = src[k];
    zbuf[i * 8 + k] = v;
    s += v * v;
  }
  raw[i] = s;
  float sq = fminf(s, BOUNDARY);          // raw >= 0 already
  inv1m[i] = 1.0f / (1.0f - sq);
}

// ---------------------------------------------------------------- main ------
// Block = one 16-row tile I of the 8192x8192 similarity matrix.
// 8 waves sweep the 512 column tiles J. Per tile:
//   C(16x16) = A(16x8) * B(8x16) via two v_wmma_f32_16x16x4_f32.
// A-layout (32-bit, 16x4): lane<16 holds K=0,1; lane>=16 holds K=2,3 for row M=lane%16.
// B-layout mirrors it; since B = Z[J-tile]^T, both operands load with the same
// per-lane pattern from zbuf.
// C-layout: lane L holds (M = r + 8*(L/16), N = L%16) in c[r].
__global__ __launch_bounds__(256) void poincare_tile_kernel(
    const float* __restrict__ zbuf, const float* __restrict__ raw,
    const float* __restrict__ inv1m, float* __restrict__ partials) {
  __shared__ float s_part[8][16];  // [wave][row-in-tile] partial denominators
  __shared__ float s_pos[8];       // [wave] positive-pair sim sums

  const int tid  = threadIdx.x;
  const int lane = tid & 31;
  const int wave = tid >> 5;
  const int I    = blockIdx.x;     // row tile
  const int nloc = lane & 15;
  const int h    = lane >> 4;
  const int kb   = h * 2;          // K base for this lane half

  // A operand (fixed for the whole block): rows I*16 + nloc, cols {kb,kb+1} / {kb+4,kb+5}
  const int rowA = I * 16 + nloc;
  v2f aLo, aHi;
  {
    const float2 lo = *(const float2*)(zbuf + rowA * 8 + kb);
    const float2 hi = *(const float2*)(zbuf + rowA * 8 + kb + 4);
    aLo[0] = lo.x; aLo[1] = lo.y;
    aHi[0] = hi.x; aHi[1] = hi.y;
  }

  // Per-lane row statistics for the 8 rows this lane's C registers cover.
  float rawI[8], invI[8];
#pragma unroll
  for (int r = 0; r < 8; ++r) {
    int m = I * 16 + 8 * h + r;
    rawI[r] = raw[m];
    invI[r] = inv1m[m];
  }

  // Which C register (if any) holds a tile-diagonal element (m == n) in this lane.
  const int  rd      = nloc - 8 * h;
  const bool hasDiag = (rd >= 0) && (rd < 8);
  const int  Jpos    = (I < NTILES / 2) ? (I + NTILES / 2) : (I - NTILES / 2);

  float acc[8];
#pragma unroll
  for (int r = 0; r < 8; ++r) acc[r] = 0.0f;
  float posAcc = 0.0f;

  for (int J = wave; J < NTILES; J += 8) {
    const int rowB = J * 16 + nloc;
    v2f bLo, bHi;
    {
      const float2 lo = *(const float2*)(zbuf + rowB * 8 + kb);
      const float2 hi = *(const float2*)(zbuf + rowB * 8 + kb + 4);
      bLo[0] = lo.x; bLo[1] = lo.y;
      bHi[0] = hi.x; bHi[1] = hi.y;
    }
    const float rawJ = raw[rowB];
    const float invJ = inv1m[rowB];

    v8f c = {};
    // D = A*B + C, f32, 16x16 tile, K=4 per instruction -> two for K=8.
    c = __builtin_amdgcn_wmma_f32_16x16x4_f32(false, aLo, false, bLo,
                                              (short)0, c, false, false);
    c = __builtin_amdgcn_wmma_f32_16x16x4_f32(false, aHi, false, bHi,
                                              (short)0, c, false, false);

#pragma unroll
    for (int r = 0; r < 8; ++r) {
      float dot  = c[r];
      float sqd  = fmaxf(rawI[r] + rawJ - 2.0f * dot, 0.0f);
      float x    = fmaf(sqd * invI[r] * invJ, 2.0f, 1.0f);    // x >= 1
      float t    = fmaxf(fmaf(x, x, -1.0f), 0.0f);
      // arcosh(x) = ln(x + sqrt(x^2-1)); u >= 1 so bare HW trans ops are safe
      float u    = x + __builtin_amdgcn_sqrtf(t);             // v_sqrt_f32
      float dist = __builtin_amdgcn_logf(u) * LN2;            // v_log_f32 (log2) * ln2
      float sim  = __builtin_amdgcn_rcpf(1.0f + dist);        // v_rcp_f32
      float e    = __builtin_amdgcn_exp2f(sim * TWO_OVER_LN2);// exp(sim/T) via v_exp_f32
      if (J == I && hasDiag && r == rd) e = 0.0f;             // exclude j == i
      acc[r] += e;
      if (J == Jpos && hasDiag && r == rd) posAcc += sim;     // positive pair
    }
  }

  // Row-sum (denominator) reduction across the 16 lanes of each half.
#pragma unroll
  for (int r = 0; r < 8; ++r) {
#pragma unroll
    for (int mask = 1; mask < 16; mask <<= 1)
      acc[r] += __shfl_xor(acc[r], mask, 32);
  }
  // Positive sum across the whole wave.
#pragma unroll
  for (int mask = 1; mask < 32; mask <<= 1)
    posAcc += __shfl_xor(posAcc, mask, 32);

  if (nloc == 0) {
#pragma unroll
    for (int r = 0; r < 8; ++r) s_part[wave][8 * h + r] = acc[r];
  }
  if (lane == 0) s_pos[wave] = posAcc;
  __syncthreads();

  // Deterministic in-order combine across waves; per-block partial loss.
  if (tid == 0) {
    float logsum = 0.0f;
    for (int m = 0; m < 16; ++m) {
      float d = 0.0f;
      for (int w = 0; w < 8; ++w) d += s_part[w][m];
      logsum += __logf(d);   // once per row: keep the precise path here
    }
    float pos = 0.0f;
    for (int w = 0; w < 8; ++w) pos += s_pos[w];
    // sum over this block's 16 rows of (log(denom) - pos/T), T = 0.5, /(2N)
    partials[I] = (logsum - 2.0f * pos) * (1.0f / 8192.0f);
  }
}

// ------------------------------------------------------------- finalize -----
__global__ void poincare_finalize_kernel(const float* __restrict__ partials,
                                         float* __restrict__ out) {
  if (blockIdx.x == 0 && threadIdx.x == 0) {
    float s = 0.0f;
    for (int i = 0; i < NTILES; ++i) s += partials[i];  // fixed order
    out[0] = s;
  }
}

// ---------------------------------------------------------------------------
extern "C" void kernel_launch(void* const* d_in, const int* in_sizes, int n_in,
                              void* d_out, int out_size, void* d_ws, size_t ws_size,
                              hipStream_t stream) {
  const float* zi = (const float*)d_in[0];   // [4096, 8] f32
  const float* zj = (const float*)d_in[1];   // [4096, 8] f32
  float* out = (float*)d_out;                // scalar f32

  // Workspace layout (floats):
  float* ws       = (float*)d_ws;
  float* zbuf     = ws;                       // 8192*8  = 65536
  float* raw      = ws + 65536;               // 8192
  float* inv1m    = ws + 73728;               // 8192
  float* partials = ws + 81920;               // 512
  (void)in_sizes; (void)n_in; (void)out_size; (void)ws_size;

  poincare_prep_kernel<<<NROWS / 256, 256, 0, stream>>>(zi, zj, zbuf, raw, inv1m);
  poincare_tile_kernel<<<NTILES, 256, 0, stream>>>(zbuf, raw, inv1m, partials);
  poincare_finalize_kernel<<<1, 32, 0, stream>>>(partials, out);
}